// BatchNeighborList_77129022701932
// MI455X (gfx1250) — compile-verified
//
#include <hip/hip_runtime.h>
#include <hip/hip_bf16.h>

typedef float v4f __attribute__((ext_vector_type(4)));
typedef int   v4i __attribute__((__vector_size__(4 * sizeof(int))));

#define CUTOFF   5.0f
#define MIN_DIST 0.01f
#define EPSW     1e-7f

#define NMAX   2048
#define BLOCK  256
#define ITILE  8     // output rows per block (amortize LDS fill: 320KB stores / 24KB fill)
#define JPT    4     // j's per thread per pass -> 128-bit stores

#if defined(__has_builtin)
#  if __has_builtin(__builtin_amdgcn_global_load_async_to_lds_b128)
#    define HAVE_ASYNC_LDS 1
#  endif
#  if __has_builtin(__builtin_amdgcn_s_wait_asynccnt)
#    define HAVE_WAIT_ASYNC 1
#  endif
#endif

#define AS_GLOBAL __attribute__((address_space(1)))
#define AS_LDS    __attribute__((address_space(3)))

__global__ __launch_bounds__(BLOCK) void BatchNeighborList_kernel(
    const float* __restrict__ pos,    // [B*N, 3]
    const float* __restrict__ cells,  // [3B, 3]
    float* __restrict__ out,          // diff[B,N,N,3] ++ dist[B,N,N] ++ mask[B,N,N]
    int B, int N)
{
    __shared__ __align__(16) float sP[NMAX * 3];   // wrapped positions of this batch

    const int b  = blockIdx.y;
    const int t  = threadIdx.x;
    const int i0 = blockIdx.x * ITILE;

    // ---- cell and its inverse (uniform; adjugate / det) ----
    const float* c = cells + b * 9;
    const float a00=c[0], a01=c[1], a02=c[2];
    const float a10=c[3], a11=c[4], a12=c[5];
    const float a20=c[6], a21=c[7], a22=c[8];
    const float m00 = a11*a22 - a12*a21;
    const float m01 = a12*a20 - a10*a22;
    const float m02 = a10*a21 - a11*a20;
    const float id  = 1.0f / (a00*m00 + a01*m01 + a02*m02);
    const float i00 =  m00*id;
    const float i01 = (a02*a21 - a01*a22)*id;
    const float i02 = (a01*a12 - a02*a11)*id;
    const float i10 =  m01*id;
    const float i11 = (a00*a22 - a02*a20)*id;
    const float i12 = (a02*a10 - a00*a12)*id;
    const float i20 =  m02*id;
    const float i21 = (a01*a20 - a00*a21)*id;
    const float i22 = (a00*a11 - a01*a10)*id;

    // ---- stage raw positions of this batch into LDS (async DMA path) ----
    const float* gsrc = pos + (size_t)b * N * 3;
    const int n16 = (N * 3) / 4;               // number of 16-byte chunks
#if HAVE_ASYNC_LDS
    for (int k = t; k < n16; k += BLOCK) {
        __builtin_amdgcn_global_load_async_to_lds_b128(
            (AS_GLOBAL v4i*)(gsrc + k * 4),
            (AS_LDS    v4i*)(&sP[k * 4]),
            0, 0);
    }
#  if HAVE_WAIT_ASYNC
    __builtin_amdgcn_s_wait_asynccnt(0);
#  else
    asm volatile("s_wait_asynccnt 0" ::: "memory");
#  endif
#else
    for (int k = t; k < n16; k += BLOCK) {
        v4f v = *(const v4f*)(gsrc + k * 4);
        *(v4f*)(&sP[k * 4]) = v;
    }
#endif
    __syncthreads();

    // ---- wrap positions in place (each element owned by one thread) ----
    for (int a = t; a < N; a += BLOCK) {
        const float p0 = sP[a*3+0], p1 = sP[a*3+1], p2 = sP[a*3+2];
        float s0 = p0*i00 + p1*i10 + p2*i20 + EPSW;
        float s1 = p0*i01 + p1*i11 + p2*i21 + EPSW;
        float s2 = p0*i02 + p1*i12 + p2*i22 + EPSW;
        s0 -= floorf(s0);  s1 -= floorf(s1);  s2 -= floorf(s2);   // % 1.0
        s0 -= EPSW;        s1 -= EPSW;        s2 -= EPSW;
        sP[a*3+0] = s0*a00 + s1*a10 + s2*a20;
        sP[a*3+1] = s0*a01 + s1*a11 + s2*a21;
        sP[a*3+2] = s0*a02 + s1*a12 + s2*a22;
    }
    __syncthreads();

    const size_t NN = (size_t)N * N;
    float* __restrict__ diffO = out;
    float* __restrict__ distO = out + (size_t)B * NN * 3;
    float* __restrict__ maskO = out + (size_t)B * NN * 4;

    for (int ii = 0; ii < ITILE; ++ii) {
        const int i = i0 + ii;
        if (i >= N) break;
        const float x0 = sP[i*3+0], x1 = sP[i*3+1], x2 = sP[i*3+2];
        const size_t rb = ((size_t)b * N + i) * N;

        for (int jb = t * JPT; jb < N; jb += BLOCK * JPT) {
            float db[JPT * 3];
            v4f dv, mv;
            #pragma unroll
            for (int k = 0; k < JPT; ++k) {
                const int j = jb + k;
                float e0 = 0.f, e1 = 0.f, e2 = 0.f, r = 0.f, mf = 0.f;
                if (j < N) {
                    const float d0 = sP[j*3+0] - x0;
                    const float d1 = sP[j*3+1] - x1;
                    const float d2 = sP[j*3+2] - x2;
                    float f0 = d0*i00 + d1*i10 + d2*i20;
                    float f1 = d0*i01 + d1*i11 + d2*i21;
                    float f2 = d0*i02 + d1*i12 + d2*i22;
                    f0 -= rintf(f0);  f1 -= rintf(f1);  f2 -= rintf(f2); // min image
                    e0 = f0*a00 + f1*a10 + f2*a20;
                    e1 = f0*a01 + f1*a11 + f2*a21;
                    e2 = f0*a02 + f1*a12 + f2*a22;
                    r  = sqrtf(e0*e0 + e1*e1 + e2*e2);
                    const bool m = (r < CUTOFF) && (r > MIN_DIST);
                    mf = m ? 1.0f : 0.0f;
                    if (!m) { e0 = 0.f; e1 = 0.f; e2 = 0.f; r = 0.f; }
                }
                db[k*3+0] = e0; db[k*3+1] = e1; db[k*3+2] = e2;
                dv[k] = r; mv[k] = mf;
            }

            if (jb + JPT <= N) {
                // streaming 128-bit stores; NT: 335MB output > 192MB L2, never re-read
                float* dp = diffO + (rb + (size_t)jb) * 3;
                v4f q0 = { db[0], db[1], db[2],  db[3]  };
                v4f q1 = { db[4], db[5], db[6],  db[7]  };
                v4f q2 = { db[8], db[9], db[10], db[11] };
                __builtin_nontemporal_store(q0, (v4f*)(dp + 0));
                __builtin_nontemporal_store(q1, (v4f*)(dp + 4));
                __builtin_nontemporal_store(q2, (v4f*)(dp + 8));
                __builtin_nontemporal_store(dv, (v4f*)(distO + rb + jb));
                __builtin_nontemporal_store(mv, (v4f*)(maskO + rb + jb));
            } else {
                for (int k = 0; k < JPT; ++k) {
                    const int j = jb + k;
                    if (j >= N) break;
                    float* dp = diffO + (rb + (size_t)j) * 3;
                    __builtin_nontemporal_store(db[k*3+0], dp + 0);
                    __builtin_nontemporal_store(db[k*3+1], dp + 1);
                    __builtin_nontemporal_store(db[k*3+2], dp + 2);
                    __builtin_nontemporal_store(dv[k], distO + rb + j);
                    __builtin_nontemporal_store(mv[k], maskO + rb + j);
                }
            }
        }
    }
}

extern "C" void kernel_launch(void* const* d_in, const int* in_sizes, int n_in,
                              void* d_out, int out_size, void* d_ws, size_t ws_size,
                              hipStream_t stream) {
    const float* pos   = (const float*)d_in[0];   // [B*N, 3]
    const float* cells = (const float*)d_in[1];   // [3B, 3]
    (void)d_ws; (void)ws_size; (void)n_in; (void)out_size;

    const int B = in_sizes[1] / 9;
    const int N = in_sizes[0] / (3 * B);
    if (N > NMAX || B <= 0) return;

    dim3 grid((N + ITILE - 1) / ITILE, B);
    BatchNeighborList_kernel<<<grid, BLOCK, 0, stream>>>(pos, cells, (float*)d_out, B, N);
}